// QGraphConvNet_2602750181802
// MI455X (gfx1250) — compile-verified
//
#include <hip/hip_runtime.h>
#include <hip/hip_bf16.h>

#ifndef __HIP_PLATFORM_AMD__
#define unsafeAtomicAdd atomicAdd
#endif

#define D_LAT 64
#define D_OUT 16
#define WP_STRIDE 66            // float2 units per (chunk,kk) group: 64 pairs + 2 pad
#define TR_STRIDE 68            // padded LDS stride for 16x64 transit tile
#define TILES_PER_BLOCK 2

typedef __attribute__((ext_vector_type(2))) float v2f;
typedef __attribute__((ext_vector_type(8))) float v8f;

__device__ __forceinline__ v8f wmma4(v2f a, v2f b, v8f c) {
  // D(16x16,f32) = A(16x4,f32) * B(4x16,f32) + C
  return __builtin_amdgcn_wmma_f32_16x16x4_f32(false, a, false, b, (short)0, c,
                                               false, false);
}

__device__ __forceinline__ v8f splat8(float v) {
  v8f r;
#pragma unroll
  for (int j = 0; j < 8; ++j) r[j] = v;
  return r;
}

// ---------------------------------------------------------------- degrees ---
__global__ __launch_bounds__(256) void init_deg_kernel(float* ds, float* dr, int n) {
  int i = blockIdx.x * 256 + threadIdx.x;
  if (i < n) { ds[i] = 1.0f; dr[i] = 1.0f; }  // self-edge contribution
}

__global__ __launch_bounds__(256) void edge_deg_kernel(const int* __restrict__ s,
                                                       const int* __restrict__ r,
                                                       float* ds, float* dr, int e) {
  int i = blockIdx.x * 256 + threadIdx.x;
  if (i < e) {
    unsafeAtomicAdd(&ds[s[i]], 1.0f);
    unsafeAtomicAdd(&dr[r[i]], 1.0f);
  }
}

__global__ __launch_bounds__(256) void finalize_deg_kernel(float* ds, float* dr, int n) {
  int i = blockIdx.x * 256 + threadIdx.x;
  if (i < n) {
    ds[i] = rsqrtf(fmaxf(ds[i], 1.0f));
    dr[i] = rsqrtf(fmaxf(dr[i], 1.0f));
  }
}

// ------------------------------------------------------------------ embed ---
__global__ __launch_bounds__(256) void embed_kernel(const float* __restrict__ nodes,
                                                    const float* __restrict__ W,
                                                    const float* __restrict__ b,
                                                    float* __restrict__ x, int n) {
  int idx = blockIdx.x * 256 + threadIdx.x;
  if (idx < n * D_LAT) {
    int i = idx >> 6, d = idx & 63;
    float n0 = nodes[i * 3 + 0], n1 = nodes[i * 3 + 1], n2 = nodes[i * 3 + 2];
    x[idx] = b[d] + n0 * W[d] + n1 * W[64 + d] + n2 * W[128 + d];
  }
}

// ------------------------------------------------- fused 2-layer MLP (WMMA) --
// h = relu(relu(x@W0+b0)@W1+b1) * inv_sqrt_s   per 16-row tile per wave.
// Weights are repacked in LDS so each WMMA B-fragment is one aligned float2:
//   sWp[(c*2+kk)*WP_STRIDE + col2] = { W[(4c+2kk)*64 + col], W[(4c+2kk+1)*64 + col] }
__global__ __launch_bounds__(64) void mlp_wmma_kernel(
    const float* __restrict__ x, const float* __restrict__ W0,
    const float* __restrict__ b0, const float* __restrict__ W1,
    const float* __restrict__ b1, const float* __restrict__ inv_s,
    float* __restrict__ h, int nTiles) {
  __shared__ float2 sW0p[32 * WP_STRIDE];
  __shared__ float2 sW1p[32 * WP_STRIDE];
  __shared__ float sB[2 * D_LAT];
  __shared__ float transit[TILES_PER_BLOCK][16 * TR_STRIDE];

  const int t = threadIdx.x;
  // cooperative weight repack: group g = c*2+kk holds K rows {4c+2kk, 4c+2kk+1}
  for (int p = t; p < 32 * 64; p += 64) {
    int g = p >> 6, e = p & 63;               // e == global column 0..63
    int c = g >> 1, kk = g & 1;
    int krow = 4 * c + 2 * kk;
    sW0p[g * WP_STRIDE + e] = make_float2(W0[krow * 64 + e], W0[(krow + 1) * 64 + e]);
    sW1p[g * WP_STRIDE + e] = make_float2(W1[krow * 64 + e], W1[(krow + 1) * 64 + e]);
  }
  sB[t] = b0[t];
  sB[64 + t] = b1[t];
  __syncthreads();

  const int wave = t >> 5, lane = t & 31;
  const int tile = blockIdx.x * TILES_PER_BLOCK + wave;
  if (tile >= nTiles) return;  // wave-uniform; EXEC stays all-1s below
  const int r0 = tile * 16;
  const int m = lane & 15;    // row (A) / column (B,C)
  const int kk = lane >> 4;   // lane-half selector
  float* tr = transit[wave];

  // stage in the 16x64 X tile (coalesced float4 global loads)
#pragma unroll
  for (int i = 0; i < 8; ++i) {
    int idx4 = i * 32 + lane;
    int row = idx4 >> 4, c4 = idx4 & 15;
    float4 v = *(const float4*)(x + (size_t)(r0 + row) * D_LAT + c4 * 4);
    *(float4*)&tr[row * TR_STRIDE + c4 * 4] = v;
  }

  // ---- layer 0 ----
  v2f a[16];
#pragma unroll
  for (int c = 0; c < 16; ++c)
    a[c] = *(const v2f*)&tr[m * TR_STRIDE + 4 * c + 2 * kk];

  v8f acc[4];
#pragma unroll
  for (int n = 0; n < 4; ++n) acc[n] = splat8(sB[16 * n + m]);

#pragma unroll
  for (int c = 0; c < 16; ++c) {
    const v2f* bp = (const v2f*)&sW0p[(c * 2 + kk) * WP_STRIDE + m];
#pragma unroll
    for (int n = 0; n < 4; ++n)
      acc[n] = wmma4(a[c], bp[n * 16], acc[n]);   // single ds_load_b64 per frag
  }
  // relu + spill to LDS (C layout -> memory layout)
#pragma unroll
  for (int n = 0; n < 4; ++n)
#pragma unroll
    for (int j = 0; j < 8; ++j)
      tr[(j + 8 * kk) * TR_STRIDE + 16 * n + m] = fmaxf(acc[n][j], 0.0f);

  // ---- layer 1 ----
#pragma unroll
  for (int c = 0; c < 16; ++c)
    a[c] = *(const v2f*)&tr[m * TR_STRIDE + 4 * c + 2 * kk];

#pragma unroll
  for (int n = 0; n < 4; ++n) acc[n] = splat8(sB[64 + 16 * n + m]);

#pragma unroll
  for (int c = 0; c < 16; ++c) {
    const v2f* bp = (const v2f*)&sW1p[(c * 2 + kk) * WP_STRIDE + m];
#pragma unroll
    for (int n = 0; n < 4; ++n)
      acc[n] = wmma4(a[c], bp[n * 16], acc[n]);
  }
#pragma unroll
  for (int n = 0; n < 4; ++n)
#pragma unroll
    for (int j = 0; j < 8; ++j)
      tr[(j + 8 * kk) * TR_STRIDE + 16 * n + m] = fmaxf(acc[n][j], 0.0f);

  // scale by inv_sqrt_s and write coalesced
#pragma unroll
  for (int i = 0; i < 8; ++i) {
    int idx4 = i * 32 + lane;
    int row = idx4 >> 4, c4 = idx4 & 15;
    float s = inv_s[r0 + row];
    float4 v = *(const float4*)&tr[row * TR_STRIDE + c4 * 4];
    v.x *= s; v.y *= s; v.z *= s; v.w *= s;
    *(float4*)(h + (size_t)(r0 + row) * D_LAT + c4 * 4) = v;
  }
}

// ---------------------------------------------------------------- scatter ---
// agg[recv] += h[send] over all real edges (self edge folded into LN kernel).
__global__ __launch_bounds__(256) void scatter_kernel(const int* __restrict__ snd,
                                                      const int* __restrict__ rcv,
                                                      const float* __restrict__ h,
                                                      float* __restrict__ agg, int e) {
  int eid = blockIdx.x * 4 + (threadIdx.x >> 6);  // wave-uniform edge id
  int f = threadIdx.x & 63;
  if (eid < e) {
    int s = snd[eid], r = rcv[eid];
    unsafeAtomicAdd(&agg[(size_t)r * D_LAT + f], h[(size_t)s * D_LAT + f]);
  }
}

// ------------------------------------------------------ skip + layer norm ---
__global__ __launch_bounds__(256) void ln_kernel(const float* __restrict__ agg,
                                                 const float* __restrict__ h,
                                                 const float* __restrict__ inv_r,
                                                 float* __restrict__ x,
                                                 const float* __restrict__ scale,
                                                 const float* __restrict__ bias, int n) {
  int wave = threadIdx.x >> 5, lane = threadIdx.x & 31;
  int row = blockIdx.x * 8 + wave;
  if (row >= n) return;
  const float ir = inv_r[row];
  size_t base = (size_t)row * D_LAT;
  // self edge: aggregated = agg (edges) + h (self), then * inv_sqrt_r, + skip
  float v0 = (agg[base + lane] + h[base + lane]) * ir + x[base + lane];
  float v1 = (agg[base + 32 + lane] + h[base + 32 + lane]) * ir + x[base + 32 + lane];
  float s = v0 + v1;
#pragma unroll
  for (int msk = 16; msk >= 1; msk >>= 1) s += __shfl_xor(s, msk, 32);
  float mu = s * (1.0f / 64.0f);
  float d0 = v0 - mu, d1 = v1 - mu;
  float q = d0 * d0 + d1 * d1;
#pragma unroll
  for (int msk = 16; msk >= 1; msk >>= 1) q += __shfl_xor(q, msk, 32);
  float rstd = rsqrtf(q * (1.0f / 64.0f) + 1e-6f);
  x[base + lane] = d0 * rstd * scale[lane] + bias[lane];
  x[base + 32 + lane] = d1 * rstd * scale[32 + lane] + bias[32 + lane];
}

// ----------------------------------------------------- pool + decode ---------
__global__ __launch_bounds__(256) void pool_decode_kernel(
    const float* __restrict__ x, const int* __restrict__ n_node,
    const float* __restrict__ Wg, const float* __restrict__ bg,
    float* __restrict__ out, int nGraph) {
  __shared__ float spart[256];
  __shared__ float spool[D_LAT];
  int g = blockIdx.x;
  int t = threadIdx.x;
  int off = 0;
  for (int i = 0; i < g; ++i) off += n_node[i];
  int cnt = n_node[g];

  int f = t & 63, rgrp = t >> 6;  // 4 row-groups x 64 features
  float acc = 0.0f;
  for (int r = rgrp; r < cnt; r += 4)
    acc += x[(size_t)(off + r) * D_LAT + f];
  spart[t] = acc;
  __syncthreads();
  if (t < D_LAT) {
    float s = spart[t] + spart[t + 64] + spart[t + 128] + spart[t + 192];
    spool[t] = s / fmaxf((float)cnt, 1.0f);
  }
  __syncthreads();
  if (t < D_OUT) {
    float o = bg[t];
#pragma unroll 8
    for (int k = 0; k < D_LAT; ++k) o += spool[k] * Wg[k * D_OUT + t];
    out[(size_t)g * D_OUT + t] = o;
  }
}

// -------------------------------------------------------------- launcher ----
extern "C" void kernel_launch(void* const* d_in, const int* in_sizes, int n_in,
                              void* d_out, int out_size, void* d_ws, size_t ws_size,
                              hipStream_t stream) {
  const float* nodes   = (const float*)d_in[0];
  const int*   senders = (const int*)d_in[1];
  const int*   recvs   = (const int*)d_in[2];
  const int*   n_node  = (const int*)d_in[3];
  const float* W_embed = (const float*)d_in[4];
  const float* b_embed = (const float*)d_in[5];
  const float* W_mlp0  = (const float*)d_in[6];
  const float* b_mlp0  = (const float*)d_in[7];
  const float* W_mlp1  = (const float*)d_in[8];
  const float* b_mlp1  = (const float*)d_in[9];
  const float* ln_s    = (const float*)d_in[10];
  const float* ln_b    = (const float*)d_in[11];
  const float* W_glob  = (const float*)d_in[12];
  const float* b_glob  = (const float*)d_in[13];
  float* out = (float*)d_out;

  const int N = in_sizes[0] / 3;
  const int E = in_sizes[1];
  const int G = in_sizes[3];

  // workspace carve-up (256B aligned)
  char* ws = (char*)d_ws;
  size_t off = 0;
  auto carve = [&](size_t bytes) {
    char* p = ws + off;
    off = (off + bytes + 255) & ~(size_t)255;
    return p;
  };
  float* x     = (float*)carve((size_t)N * D_LAT * sizeof(float));
  float* h     = (float*)carve((size_t)N * D_LAT * sizeof(float));
  float* agg   = (float*)carve((size_t)N * D_LAT * sizeof(float));
  float* inv_s = (float*)carve((size_t)N * sizeof(float));
  float* inv_r = (float*)carve((size_t)N * sizeof(float));
  (void)ws_size;

  const int nBlkN   = (N + 255) / 256;
  const int nBlkE   = (E + 255) / 256;
  const int nTiles  = (N + 15) / 16;
  const int nBlkMLP = (nTiles + TILES_PER_BLOCK - 1) / TILES_PER_BLOCK;

  // degrees -> inv_sqrt (self edge included via init=1)
  init_deg_kernel<<<nBlkN, 256, 0, stream>>>(inv_s, inv_r, N);
  edge_deg_kernel<<<nBlkE, 256, 0, stream>>>(senders, recvs, inv_s, inv_r, E);
  finalize_deg_kernel<<<nBlkN, 256, 0, stream>>>(inv_s, inv_r, N);

  // embed
  embed_kernel<<<(N * D_LAT + 255) / 256, 256, 0, stream>>>(nodes, W_embed, b_embed, x, N);

  for (int step = 0; step < 3; ++step) {
    hipMemsetAsync(agg, 0, (size_t)N * D_LAT * sizeof(float), stream);
    mlp_wmma_kernel<<<nBlkMLP, 64, 0, stream>>>(x, W_mlp0, b_mlp0, W_mlp1, b_mlp1,
                                                inv_s, h, nTiles);
    scatter_kernel<<<(E + 3) / 4, 256, 0, stream>>>(senders, recvs, h, agg, E);
    ln_kernel<<<(N + 7) / 8, 256, 0, stream>>>(agg, h, inv_r, x, ln_s, ln_b, N);
  }

  pool_decode_kernel<<<G, 256, 0, stream>>>(x, n_node, W_glob, b_glob, out, G);
}